// MultiHeadAttention_1580547971887
// MI455X (gfx1250) — compile-verified
//
#include <hip/hip_runtime.h>
#include <stdint.h>

// MI455X / gfx1250: wave32, WMMA f32 16x16x4 for full-precision fp32 GEMMs.
// GEMM kernels are one-wave-per-block with __launch_bounds__(32,1) so the
// 128-VGPR accumulator tile stays register-resident (no scratch spills).

typedef __attribute__((ext_vector_type(2))) float v2f;
typedef __attribute__((ext_vector_type(8))) float v8f;

constexpr int Bc = 8;
constexpr int Cc = 1024;
constexpr int Tc = 1024;
constexpr int Hh = 16;
constexpr int Dd = 64;            // C / H
constexpr float GN_EPS = 1e-5f;
constexpr float NEGV = -1e30f;
constexpr float SCALE = 0.125f;   // 1/sqrt(64)

__device__ __forceinline__ v8f wmma4(v2f a, v2f b, v8f c) {
  // V_WMMA_F32_16X16X4_F32: D = A(16x4) * B(4x16) + C(16x16)
  return __builtin_amdgcn_wmma_f32_16x16x4_f32(
      false, a, false, b, (short)0, c, false, false);
}

// ---------------------------------------------------------------------------
// Kernel 1: kvq[b,o,t] = sum_c w_kvq[o,c] * x[b,c,t]   (o in [0,3C))
// One wave per 64x64 output tile, K-loop over C in steps of 4.
// ---------------------------------------------------------------------------
__global__ void __launch_bounds__(32, 1)
k_qkv(const float* __restrict__ W, const float* __restrict__ X,
      float* __restrict__ KVQ) {
  const int lane = threadIdx.x;
  const int half = lane >> 4;
  const int n    = lane & 15;
  const int mBase = blockIdx.x * 64;  // over 3C
  const int nBase = blockIdx.y * 64;  // over T
  const int b     = blockIdx.z;
  const float* Xb = X + (size_t)b * Cc * Tc;
  float* Ob = KVQ + (size_t)b * 3 * Cc * Tc;

  v8f acc[4][4] = {};
  for (int c0 = 0; c0 < Cc; c0 += 4) {
    v2f a[4], bb[4];
#pragma unroll
    for (int mi = 0; mi < 4; ++mi) {
      // A[m][k] = W[mBase+m][c0+k]; lane layout: K = 2*half + {0,1}
      const float* p = W + (size_t)(mBase + mi * 16 + n) * Cc + c0 + 2 * half;
      a[mi] = *(const v2f*)p;
    }
#pragma unroll
    for (int ni = 0; ni < 4; ++ni) {
      // B[k][col] = X[c0+k][nBase+col]
      const int col = nBase + ni * 16 + n;
      bb[ni].x = Xb[(size_t)(c0 + 2 * half) * Tc + col];
      bb[ni].y = Xb[(size_t)(c0 + 2 * half + 1) * Tc + col];
    }
#pragma unroll
    for (int mi = 0; mi < 4; ++mi)
#pragma unroll
      for (int ni = 0; ni < 4; ++ni)
        acc[mi][ni] = wmma4(a[mi], bb[ni], acc[mi][ni]);
  }
#pragma unroll
  for (int mi = 0; mi < 4; ++mi)
#pragma unroll
    for (int ni = 0; ni < 4; ++ni)
#pragma unroll
      for (int r = 0; r < 8; ++r) {
        const int row = mBase + mi * 16 + half * 8 + r;
        const int col = nBase + ni * 16 + n;
        Ob[(size_t)row * Tc + col] = acc[mi][ni][r];
      }
}

// ---------------------------------------------------------------------------
// Kernel 2: att[z, tk, tq] = maskK[tk] ? NEG : scale * sum_d K[d,tk]*Q[d,tq]
// z = h*B + b (torch return layout). Written directly into d_out att region.
// ---------------------------------------------------------------------------
__global__ void __launch_bounds__(32, 1)
k_att(const float* __restrict__ KVQ, const uint8_t* __restrict__ mask,
      float* __restrict__ ATT) {
  const int lane = threadIdx.x;
  const int half = lane >> 4;
  const int n    = lane & 15;
  const int tkBase = blockIdx.x * 64;
  const int tqBase = blockIdx.y * 64;
  const int z = blockIdx.z;            // z = h*8 + b  ->  b = z&7, h = z>>3
  const int b = z & 7;
  const int h = z >> 3;
  const float* Kb = KVQ + ((size_t)b * 3 * Cc + (size_t)h * Dd) * Tc;
  const float* Qb = KVQ + ((size_t)b * 3 * Cc + 2 * Cc + (size_t)h * Dd) * Tc;
  float* S = ATT + (size_t)z * Tc * Tc;

  v8f acc[4][4] = {};
#pragma unroll 2
  for (int d0 = 0; d0 < Dd; d0 += 4) {
    v2f a[4], bb[4];
#pragma unroll
    for (int mi = 0; mi < 4; ++mi) {
      // A = K^T: A[m][k] = K[d0+k][tkBase+m]
      const int row = tkBase + mi * 16 + n;
      a[mi].x = Kb[(size_t)(d0 + 2 * half) * Tc + row];
      a[mi].y = Kb[(size_t)(d0 + 2 * half + 1) * Tc + row];
    }
#pragma unroll
    for (int ni = 0; ni < 4; ++ni) {
      const int col = tqBase + ni * 16 + n;
      bb[ni].x = Qb[(size_t)(d0 + 2 * half) * Tc + col];
      bb[ni].y = Qb[(size_t)(d0 + 2 * half + 1) * Tc + col];
    }
#pragma unroll
    for (int mi = 0; mi < 4; ++mi)
#pragma unroll
      for (int ni = 0; ni < 4; ++ni)
        acc[mi][ni] = wmma4(a[mi], bb[ni], acc[mi][ni]);
  }
#pragma unroll
  for (int mi = 0; mi < 4; ++mi) {
#pragma unroll
    for (int r = 0; r < 8; ++r) {
      const int tk = tkBase + mi * 16 + half * 8 + r;
      const float keep = mask[b * Tc + tk] ? 0.0f : 1.0f;
#pragma unroll
      for (int ni = 0; ni < 4; ++ni) {
        const int tq = tqBase + ni * 16 + n;
        const float v = acc[mi][ni][r] * SCALE;
        S[(size_t)tk * Tc + tq] = keep * v + (1.0f - keep) * NEGV;
      }
    }
  }
}

// ---------------------------------------------------------------------------
// Kernel 3: softmax over tk (per column tq), then zero masked query columns.
// One thread per column -> fully coalesced row traffic; slabs are L2-hot.
// ---------------------------------------------------------------------------
__global__ void k_softmax(float* __restrict__ ATT,
                          const uint8_t* __restrict__ mask) {
  const int z = blockIdx.y;
  const int b = z & 7;
  const int tq = blockIdx.x * blockDim.x + threadIdx.x;
  float* S = ATT + (size_t)z * Tc * Tc;

  float m = -INFINITY, sum = 0.0f;
  for (int tk = 0; tk < Tc; ++tk) {
    const float v = S[(size_t)tk * Tc + tq];
    const float nm = fmaxf(m, v);
    sum = sum * __expf(m - nm) + __expf(v - nm);
    m = nm;
  }
  const float inv = 1.0f / sum;
  const float qkeep = mask[b * Tc + tq] ? 0.0f : 1.0f;
  for (int tk = 0; tk < Tc; ++tk) {
    const size_t idx = (size_t)tk * Tc + tq;
    S[idx] = __expf(S[idx] - m) * inv * qkeep;
  }
}

// ---------------------------------------------------------------------------
// Kernel 4: ctx[b, h*64+dd, tq] = sum_tk V[dd,tk] * att[z,tk,tq]
// One wave per 64(d) x 64(tq) tile; K-loop over T.
// ---------------------------------------------------------------------------
__global__ void __launch_bounds__(32, 1)
k_ctx(const float* __restrict__ KVQ, const float* __restrict__ ATT,
      float* __restrict__ CTX) {
  const int lane = threadIdx.x;
  const int half = lane >> 4;
  const int n    = lane & 15;
  const int tqBase = blockIdx.x * 64;
  const int z = blockIdx.y;
  const int b = z & 7;
  const int h = z >> 3;
  const float* Vb = KVQ + ((size_t)b * 3 * Cc + Cc + (size_t)h * Dd) * Tc;
  const float* S  = ATT + (size_t)z * Tc * Tc;
  float* Ob = CTX + ((size_t)b * Cc + (size_t)h * Dd) * Tc;

  v8f acc[4][4] = {};
  for (int k0 = 0; k0 < Tc; k0 += 4) {
    v2f a[4], bb[4];
#pragma unroll
    for (int mi = 0; mi < 4; ++mi) {
      // A = V: A[m][k] = V[m][k0+k]
      const float* p = Vb + (size_t)(mi * 16 + n) * Tc + k0 + 2 * half;
      a[mi] = *(const v2f*)p;
    }
#pragma unroll
    for (int ni = 0; ni < 4; ++ni) {
      const int col = tqBase + ni * 16 + n;
      bb[ni].x = S[(size_t)(k0 + 2 * half) * Tc + col];
      bb[ni].y = S[(size_t)(k0 + 2 * half + 1) * Tc + col];
    }
#pragma unroll
    for (int mi = 0; mi < 4; ++mi)
#pragma unroll
      for (int ni = 0; ni < 4; ++ni)
        acc[mi][ni] = wmma4(a[mi], bb[ni], acc[mi][ni]);
  }
#pragma unroll
  for (int mi = 0; mi < 4; ++mi)
#pragma unroll
    for (int ni = 0; ni < 4; ++ni)
#pragma unroll
      for (int r = 0; r < 8; ++r) {
        const int dd  = mi * 16 + half * 8 + r;
        const int col = tqBase + ni * 16 + n;
        Ob[(size_t)dd * Tc + col] = acc[mi][ni][r];
      }
}

// ---------------------------------------------------------------------------
// Kernel 5: ypre[b,o,t] = sum_c w_out[o,c]*ctx[b,c,t] + x[b,o,t]
// ypre stored into the (dead) K rows of the kvq workspace: row (b*3C + o).
// ---------------------------------------------------------------------------
__global__ void __launch_bounds__(32, 1)
k_out(const float* __restrict__ W, const float* __restrict__ CTX,
      const float* __restrict__ X, float* __restrict__ YP) {
  const int lane = threadIdx.x;
  const int half = lane >> 4;
  const int n    = lane & 15;
  const int mBase = blockIdx.x * 64;  // over C
  const int nBase = blockIdx.y * 64;  // over T
  const int b     = blockIdx.z;
  const float* Cb = CTX + (size_t)b * Cc * Tc;
  const float* Xb = X + (size_t)b * Cc * Tc;
  float* Ob = YP + (size_t)b * 3 * Cc * Tc;  // strided reuse of kvq K region

  v8f acc[4][4] = {};
  for (int c0 = 0; c0 < Cc; c0 += 4) {
    v2f a[4], bb[4];
#pragma unroll
    for (int mi = 0; mi < 4; ++mi) {
      const float* p = W + (size_t)(mBase + mi * 16 + n) * Cc + c0 + 2 * half;
      a[mi] = *(const v2f*)p;
    }
#pragma unroll
    for (int ni = 0; ni < 4; ++ni) {
      const int col = nBase + ni * 16 + n;
      bb[ni].x = Cb[(size_t)(c0 + 2 * half) * Tc + col];
      bb[ni].y = Cb[(size_t)(c0 + 2 * half + 1) * Tc + col];
    }
#pragma unroll
    for (int mi = 0; mi < 4; ++mi)
#pragma unroll
      for (int ni = 0; ni < 4; ++ni)
        acc[mi][ni] = wmma4(a[mi], bb[ni], acc[mi][ni]);
  }
#pragma unroll
  for (int mi = 0; mi < 4; ++mi)
#pragma unroll
    for (int ni = 0; ni < 4; ++ni)
#pragma unroll
      for (int r = 0; r < 8; ++r) {
        const int row = mBase + mi * 16 + half * 8 + r;
        const int col = nBase + ni * 16 + n;
        Ob[(size_t)row * Tc + col] =
            acc[mi][ni][r] + Xb[(size_t)row * Tc + col];
      }
}

// ---------------------------------------------------------------------------
// GroupNorm(1, C): per-batch mean/var over (C,T) in double, then normalize.
// ---------------------------------------------------------------------------
__global__ void k_zero_stats(double* __restrict__ st) {
  if (threadIdx.x < 2 * Bc) st[threadIdx.x] = 0.0;
}

__global__ void k_stats(const float* __restrict__ YP, double* __restrict__ st) {
  __shared__ double s1[256];
  __shared__ double s2[256];
  const int b = blockIdx.y;
  const int tid = threadIdx.x;
  const float* Yb = YP + (size_t)b * 3 * Cc * Tc;  // rows 0..C-1 contiguous
  const size_t base = (size_t)blockIdx.x * 1024;
  float ls = 0.0f, lq = 0.0f;
#pragma unroll
  for (int k = 0; k < 4; ++k) {
    const float v = Yb[base + (size_t)k * 256 + tid];
    ls += v;
    lq += v * v;
  }
  s1[tid] = (double)ls;
  s2[tid] = (double)lq;
  __syncthreads();
  for (int s = 128; s > 0; s >>= 1) {
    if (tid < s) {
      s1[tid] += s1[tid + s];
      s2[tid] += s2[tid + s];
    }
    __syncthreads();
  }
  if (tid == 0) {
    atomicAdd(&st[b * 2 + 0], s1[0]);
    atomicAdd(&st[b * 2 + 1], s2[0]);
  }
}

__global__ void k_norm(const float* __restrict__ YP,
                       const double* __restrict__ st,
                       const float* __restrict__ gw,
                       const float* __restrict__ gb,
                       float* __restrict__ Y) {
  const size_t e = (size_t)blockIdx.x * blockDim.x + threadIdx.x;
  const int b = (int)(e / ((size_t)Cc * Tc));
  const size_t r = e % ((size_t)Cc * Tc);
  const int c = (int)(r / Tc);
  const double N = (double)Cc * (double)Tc;
  const double mean = st[b * 2 + 0] / N;
  const double var = st[b * 2 + 1] / N - mean * mean;
  const float inv = rsqrtf((float)var + GN_EPS);
  const float v = YP[(size_t)b * 3 * Cc * Tc + r];
  Y[e] = (v - (float)mean) * inv * gw[c] + gb[c];
}

// ---------------------------------------------------------------------------
// Launch
// ---------------------------------------------------------------------------
extern "C" void kernel_launch(void* const* d_in, const int* in_sizes, int n_in,
                              void* d_out, int out_size, void* d_ws,
                              size_t ws_size, hipStream_t stream) {
  const float*   x     = (const float*)d_in[0];
  const uint8_t* mask  = (const uint8_t*)d_in[1];
  const float*   w_kvq = (const float*)d_in[2];
  const float*   w_out = (const float*)d_in[3];
  const float*   gnw   = (const float*)d_in[4];
  const float*   gnb   = (const float*)d_in[5];

  float* y   = (float*)d_out;                         // [B, C, T]
  float* att = y + (size_t)Bc * Cc * Tc;              // [H*B, T, T]

  // Workspace: kvq 96MB | ctx 32MB | stats 128B  (ypre reuses kvq K rows)
  float*  kvq = (float*)d_ws;
  float*  ctx = (float*)((char*)d_ws + (size_t)Bc * 3 * Cc * Tc * 4);
  double* st  = (double*)((char*)d_ws + (size_t)Bc * 4 * Cc * Tc * 4);
  float*  ypre = kvq;

  k_qkv<<<dim3(3 * Cc / 64, Tc / 64, Bc), 32, 0, stream>>>(w_kvq, x, kvq);
  k_att<<<dim3(Tc / 64, Tc / 64, Hh * Bc), 32, 0, stream>>>(kvq, mask, att);
  k_softmax<<<dim3(Tc / 256, Hh * Bc), 256, 0, stream>>>(att, mask);
  k_ctx<<<dim3(Tc / 64, Hh * Bc), 32, 0, stream>>>(kvq, att, ctx);
  k_out<<<dim3(Cc / 64, Tc / 64, Bc), 32, 0, stream>>>(w_out, ctx, x, ypre);
  k_zero_stats<<<1, 32, 0, stream>>>(st);
  k_stats<<<dim3(Cc * Tc / 1024, Bc), 256, 0, stream>>>(ypre, st);
  k_norm<<<dim3((unsigned)((size_t)Bc * Cc * Tc / 256)), 256, 0, stream>>>(
      ypre, st, gnw, gnb, y);
}